// GATv2TrajectoryPredictor_67070209295088
// MI455X (gfx1250) — compile-verified
//
#include <hip/hip_runtime.h>
#include <cstdint>
#include <cstddef>

typedef __attribute__((ext_vector_type(2))) float v2f;
typedef __attribute__((ext_vector_type(8))) float v8f;

#define NEG_SLOPE 0.2f

// ---------------------------------------------------------------------------
// Exact-fp32 WMMA GEMM:  C[M,N] = A[M,K] @ W[K,N] + bias[N]
// One wave (32 lanes) per 16x16 output tile, K stepped by 4 via
// V_WMMA_F32_16X16X4_F32 (fp32 in, fp32 accumulate -> matches reference).
// A 16x4 layout: lanes 0-15 hold rows M=0..15 with K=k,k+1 in VGPR0/1;
//                lanes 16-31 hold the same rows with K=k+2,k+3.
// B 4x16 layout mirrors A (col striped across lanes).
// C/D: v[i] = (row = i + 8*(lane>>4), col = lane&15).
// ---------------------------------------------------------------------------
__global__ void gemm_bias_f32_wmma(const float* __restrict__ A,
                                   const float* __restrict__ W,
                                   const float* __restrict__ bias,
                                   float* __restrict__ C,
                                   int M, int K, int N) {
  const int lane  = threadIdx.x & 31;
  const int tileM = blockIdx.x;
  const int tileN = blockIdx.y;
  const int arow  = tileM * 16 + (lane & 15);
  const int col   = tileN * 16 + (lane & 15);
  const int kOff  = (lane >> 4) * 2;

  v8f acc = {};
  for (int k = 0; k < K; k += 4) {
    v2f a, b;
    a.x = A[(size_t)arow * K + (k + kOff)];
    a.y = A[(size_t)arow * K + (k + kOff + 1)];
    b.x = W[(size_t)(k + kOff)     * N + col];
    b.y = W[(size_t)(k + kOff + 1) * N + col];
    acc = __builtin_amdgcn_wmma_f32_16x16x4_f32(false, a, false, b,
                                                (short)0, acc, false, false);
  }
  const float bv    = bias[col];
  const int   rbase = tileM * 16 + (lane >> 4) * 8;
#pragma unroll
  for (int i = 0; i < 8; ++i)
    C[(size_t)(rbase + i) * N + col] = acc[i] + bv;
}

// ---------------------------------------------------------------------------
// Float atomic max via sign-aware integer atomics (exact, no CAS loop).
// Valid when the location is initialized to -INF.
// ---------------------------------------------------------------------------
__device__ __forceinline__ void atomicMaxFloat(float* addr, float val) {
  if (val >= 0.0f)
    atomicMax(reinterpret_cast<int*>(addr), __float_as_int(val));
  else
    atomicMin(reinterpret_cast<unsigned int*>(addr), __float_as_uint(val));
}

// Init m = -inf, s = 0, agg = 0
__global__ void init_buffers(float* __restrict__ m, float* __restrict__ s,
                             float* __restrict__ agg, int nm, int nagg) {
  int i = blockIdx.x * blockDim.x + threadIdx.x;
  if (i < nm)  { m[i] = -INFINITY; s[i] = 0.0f; }
  if (i < nagg) agg[i] = 0.0f;
}

// ---------------------------------------------------------------------------
// Pass 1: wave-per-edge. score[e,h] = sum_c att[h,c]*leaky_relu(xl[src]+xr[dst])
// and atomic segment-max into mmax[dst,h].
// Edges e >= E_real are self-loops: src = dst = e - E_real.
// ---------------------------------------------------------------------------
__global__ void edge_score_max(const float* __restrict__ xl,
                               const float* __restrict__ xr,
                               const long long* __restrict__ esrc,
                               const long long* __restrict__ edst,
                               const float* __restrict__ att,
                               float* __restrict__ score,
                               float* __restrict__ mmax,
                               int E_real, int E_tot, int H, int Cc) {
  const int wid  = blockIdx.x * (blockDim.x >> 5) + (threadIdx.x >> 5);
  const int lane = threadIdx.x & 31;
  if (wid >= E_tot) return;
  long long s, d;
  if (wid < E_real) { s = esrc[wid]; d = edst[wid]; }
  else              { s = wid - E_real; d = s; }
  const int dim = H * Cc;
  const float* xls = xl + (size_t)s * dim;
  const float* xrd = xr + (size_t)d * dim;
  for (int h = 0; h < H; ++h) {
    float acc = 0.0f;
    for (int c = lane; c < Cc; c += 32) {
      float v = xls[h * Cc + c] + xrd[h * Cc + c];
      v = (v > 0.0f) ? v : NEG_SLOPE * v;
      acc += att[h * Cc + c] * v;
    }
#pragma unroll
    for (int off = 16; off > 0; off >>= 1)
      acc += __shfl_xor(acc, off, 32);
    if (lane == 0) {
      score[(size_t)wid * H + h] = acc;
      atomicMaxFloat(&mmax[(size_t)d * H + h], acc);
    }
  }
}

// Pass 2: p = exp(score - m[dst]); overwrite score with p; s[dst] += p
__global__ void edge_softmax_num(const long long* __restrict__ edst,
                                 float* __restrict__ score,
                                 const float* __restrict__ mmax,
                                 float* __restrict__ ssum,
                                 int E_real, int E_tot, int H) {
  int i = blockIdx.x * blockDim.x + threadIdx.x;
  if (i >= E_tot * H) return;
  int e = i / H, h = i - e * H;
  long long d = (e < E_real) ? edst[e] : (long long)(e - E_real);
  float p = expf(score[i] - mmax[(size_t)d * H + h]);
  score[i] = p;
  atomicAdd(&ssum[(size_t)d * H + h], p);
}

// Pass 3: wave-per-edge scatter-add: agg[dst,:] += alpha * xl[src,:]
__global__ void edge_aggregate(const float* __restrict__ xl,
                               const long long* __restrict__ esrc,
                               const long long* __restrict__ edst,
                               const float* __restrict__ p,
                               const float* __restrict__ ssum,
                               float* __restrict__ agg,
                               int E_real, int E_tot, int H, int Cc) {
  const int wid  = blockIdx.x * (blockDim.x >> 5) + (threadIdx.x >> 5);
  const int lane = threadIdx.x & 31;
  if (wid >= E_tot) return;
  long long s, d;
  if (wid < E_real) { s = esrc[wid]; d = edst[wid]; }
  else              { s = wid - E_real; d = s; }
  const int dim = H * Cc;
  const float* xls = xl + (size_t)s * dim;
  float*       agd = agg + (size_t)d * dim;
  for (int idx = lane; idx < dim; idx += 32) {
    int h = idx / Cc;
    float alpha = p[(size_t)wid * H + h] /
                  (ssum[(size_t)d * H + h] + 1e-16f);
    atomicAdd(&agd[idx], alpha * xls[idx]);
  }
}

// out = (elu ? elu(agg + b) : agg + b)   (H=1 mean is identity)
__global__ void bias_act(const float* __restrict__ agg,
                         const float* __restrict__ b,
                         float* __restrict__ out,
                         int total, int dim, int apply_elu) {
  int i = blockIdx.x * blockDim.x + threadIdx.x;
  if (i >= total) return;
  int j = i % dim;
  float v = agg[i] + b[j];
  if (apply_elu) v = (v > 0.0f) ? v : expm1f(v);
  out[i] = v;
}

// Final fc: out[n,o] = h[n,:] @ Wfc + bfc   (K=64, O=2) — fully writes d_out
__global__ void fc_kernel(const float* __restrict__ h,
                          const float* __restrict__ W,
                          const float* __restrict__ b,
                          float* __restrict__ out,
                          int N, int K, int O) {
  int n = blockIdx.x * blockDim.x + threadIdx.x;
  if (n >= N) return;
  for (int o = 0; o < O; ++o) {
    float acc = b[o];
    for (int k = 0; k < K; ++k)
      acc += h[(size_t)n * K + k] * W[k * O + o];
    out[(size_t)n * O + o] = acc;
  }
}

// ---------------------------------------------------------------------------
extern "C" void kernel_launch(void* const* d_in, const int* in_sizes, int n_in,
                              void* d_out, int out_size, void* d_ws, size_t ws_size,
                              hipStream_t stream) {
  const float*     x  = (const float*)d_in[0];
  const long long* ei = (const long long*)d_in[1];   // int64 edge_index [2,E]

  const int N  = in_sizes[0] / 16;   // 30000
  const int E  = in_sizes[1] / 2;    // 480000
  const int ET = E + N;              // + self-loops
  const long long* esrc = ei;
  const long long* edst = ei + E;

  // workspace carve-up (floats)
  float* ws  = (float*)d_ws;
  size_t off = 0;
  float* XL  = ws + off; off += (size_t)N * 256;
  float* XR  = ws + off; off += (size_t)N * 256;
  float* AG  = ws + off; off += (size_t)N * 256;
  float* Hb  = ws + off; off += (size_t)N * 256;
  float* SC  = ws + off; off += (size_t)ET * 4;
  float* Mm  = ws + off; off += (size_t)N * 4;
  float* Ss  = ws + off; off += (size_t)N * 4;

  const int edgeBlocks = (ET + 7) / 8;   // 8 waves/block, wave-per-edge

  auto layer = [&](const float* X, int Kin,
                   const float* Wl, const float* bl,
                   const float* Wr, const float* br,
                   const float* att, const float* b,
                   int H, int C, int elu) {
    const int dim = H * C;
    dim3 g(N / 16, dim / 16);
    gemm_bias_f32_wmma<<<g, 32, 0, stream>>>(X, Wl, bl, XL, N, Kin, dim);
    gemm_bias_f32_wmma<<<g, 32, 0, stream>>>(X, Wr, br, XR, N, Kin, dim);
    const int tot = N * dim;
    init_buffers<<<(tot + 255) / 256, 256, 0, stream>>>(Mm, Ss, AG, N * H, tot);
    edge_score_max<<<edgeBlocks, 256, 0, stream>>>(XL, XR, esrc, edst, att,
                                                   SC, Mm, E, ET, H, C);
    edge_softmax_num<<<((size_t)ET * H + 255) / 256, 256, 0, stream>>>(
        edst, SC, Mm, Ss, E, ET, H);
    edge_aggregate<<<edgeBlocks, 256, 0, stream>>>(XL, esrc, edst, SC, Ss, AG,
                                                   E, ET, H, C);
    bias_act<<<(tot + 255) / 256, 256, 0, stream>>>(AG, b, Hb, tot, dim, elu);
  };

  // conv1: 16 -> 4x64 concat, ELU
  layer(x, 16,
        (const float*)d_in[2], (const float*)d_in[3],
        (const float*)d_in[4], (const float*)d_in[5],
        (const float*)d_in[6], (const float*)d_in[7], 4, 64, 1);
  // conv2: 256 -> 4x64 concat, ELU
  layer(Hb, 256,
        (const float*)d_in[8],  (const float*)d_in[9],
        (const float*)d_in[10], (const float*)d_in[11],
        (const float*)d_in[12], (const float*)d_in[13], 4, 64, 1);
  // conv3: 256 -> 1x64 (mean over 1 head == identity), no ELU
  layer(Hb, 256,
        (const float*)d_in[14], (const float*)d_in[15],
        (const float*)d_in[16], (const float*)d_in[17],
        (const float*)d_in[18], (const float*)d_in[19], 1, 64, 0);
  // fc: 64 -> 2
  fc_kernel<<<(N + 255) / 256, 256, 0, stream>>>(
      Hb, (const float*)d_in[20], (const float*)d_in[21],
      (float*)d_out, N, 64, 2);
}